// GIN_50886772523363
// MI455X (gfx1250) — compile-verified
//
#include <hip/hip_runtime.h>

typedef __attribute__((ext_vector_type(2))) float v2f;
typedef __attribute__((ext_vector_type(8))) float v8f;

#define N_FEAT 128

// ---------------------------------------------------------------------------
// Edge scatter-add: feat_dst[dst[e]][:] += feat_src[src[e]][:]  (F = 128)
// One wave (32 lanes) per edge, 4 floats per lane -> coalesced 512B gather,
// global_atomic_add_f32 RMW at L2.
// ---------------------------------------------------------------------------
__global__ void gin_scatter_add(const float* __restrict__ feat_src,
                                float* __restrict__ feat_dst,
                                const int* __restrict__ src_idx,
                                const int* __restrict__ dst_idx,
                                int n_edges) {
  long long gid = (long long)blockIdx.x * blockDim.x + threadIdx.x;
  int edge = (int)(gid >> 5);
  int lane = (int)(gid & 31);
  if (edge >= n_edges) return;
  int s = src_idx[edge];
  int d = dst_idx[edge];
  const float4 v = *(const float4*)(feat_src + (size_t)s * N_FEAT + lane * 4);
  float* o = feat_dst + (size_t)d * N_FEAT + lane * 4;
  atomicAdd(o + 0, v.x);
  atomicAdd(o + 1, v.y);
  atomicAdd(o + 2, v.z);
  atomicAdd(o + 3, v.w);
}

// ---------------------------------------------------------------------------
// Fused GIN MLP: out = relu( relu(hin @ Wa + ba) @ Wb + bb )
// fp32 WMMA 16x16x4. Block = (NHID/16) waves; wave w owns output cols
// [16w, 16w+16). Stage-1 activations exchanged via LDS (K-dim of stage 2).
// Weight B-fragments preloaded into registers, reused across node tiles.
// Requires n_nodes % 16 == 0 (100000 = 6250 * 16).
// ---------------------------------------------------------------------------
template <int KIN, int NHID, int NOUT>
__global__ void __launch_bounds__(2 * NHID)
gin_mlp(const float* __restrict__ hin,
        const float* __restrict__ Wa, const float* __restrict__ ba,
        const float* __restrict__ Wb, const float* __restrict__ bb,
        float* __restrict__ out, int n_nodes) {
  __shared__ float lds_a[16 * KIN];   // input tile  (16 rows x KIN)
  __shared__ float lds_t[16 * NHID];  // hidden tile (16 rows x NHID)

  const int tid   = threadIdx.x;
  const int lane  = tid & 31;
  const int wave  = tid >> 5;
  const int lhalf = lane >> 4;   // 0: lanes 0-15, 1: lanes 16-31
  const int lmod  = lane & 15;
  const int ncol  = wave * 16 + lmod;   // this wave's output column

  // --- preload B-matrix fragments (B 4x16: VGPR v, lane half selects K pair)
  v2f wa[KIN / 4];
#pragma unroll
  for (int kt = 0; kt < KIN / 4; ++kt) {
    int k = 4 * kt + 2 * lhalf;
    wa[kt].x = Wa[(size_t)k * NHID + ncol];
    wa[kt].y = Wa[(size_t)(k + 1) * NHID + ncol];
  }
  v2f wb[NHID / 4];
#pragma unroll
  for (int kt = 0; kt < NHID / 4; ++kt) {
    int k = 4 * kt + 2 * lhalf;
    wb[kt].x = Wb[(size_t)k * NOUT + ncol];
    wb[kt].y = Wb[(size_t)(k + 1) * NOUT + ncol];
  }
  const float biasa = ba[ncol];
  const float biasb = bb[ncol];

  const int ntiles = n_nodes / 16;
  for (int tile = blockIdx.x; tile < ntiles; tile += gridDim.x) {
    const int row0 = tile * 16;

    // cooperative, coalesced A-tile load
    for (int i = tid * 4; i < 16 * KIN; i += blockDim.x * 4) {
      *(float4*)&lds_a[i] = *(const float4*)&hin[(size_t)row0 * KIN + i];
    }
    __syncthreads();

    // ---- stage 1: t = relu(A @ Wa + ba) ----
    v8f acc0 = {}; v8f acc1 = {};
#pragma unroll
    for (int kt = 0; kt < KIN / 4; kt += 2) {
      {
        int k = 4 * kt + 2 * lhalf;
        v2f a; a.x = lds_a[lmod * KIN + k]; a.y = lds_a[lmod * KIN + k + 1];
        acc0 = __builtin_amdgcn_wmma_f32_16x16x4_f32(false, a, false, wa[kt],
                                                     (short)0, acc0, false, false);
      }
      {
        int k = 4 * (kt + 1) + 2 * lhalf;
        v2f a; a.x = lds_a[lmod * KIN + k]; a.y = lds_a[lmod * KIN + k + 1];
        acc1 = __builtin_amdgcn_wmma_f32_16x16x4_f32(false, a, false, wa[kt + 1],
                                                     (short)0, acc1, false, false);
      }
    }
#pragma unroll
    for (int g = 0; g < 8; ++g) {
      float v = acc0[g] + acc1[g] + biasa;      // C/D: row = g + 8*lhalf, col = ncol
      v = v > 0.0f ? v : 0.0f;
      lds_t[(g + 8 * lhalf) * NHID + ncol] = v;
    }
    __syncthreads();

    // ---- stage 2: out = relu(t @ Wb + bb) ----
    v8f acc2 = {}; v8f acc3 = {};
#pragma unroll
    for (int kt = 0; kt < NHID / 4; kt += 2) {
      {
        int k = 4 * kt + 2 * lhalf;
        v2f a; a.x = lds_t[lmod * NHID + k]; a.y = lds_t[lmod * NHID + k + 1];
        acc2 = __builtin_amdgcn_wmma_f32_16x16x4_f32(false, a, false, wb[kt],
                                                     (short)0, acc2, false, false);
      }
      {
        int k = 4 * (kt + 1) + 2 * lhalf;
        v2f a; a.x = lds_t[lmod * NHID + k]; a.y = lds_t[lmod * NHID + k + 1];
        acc3 = __builtin_amdgcn_wmma_f32_16x16x4_f32(false, a, false, wb[kt + 1],
                                                     (short)0, acc3, false, false);
      }
    }
#pragma unroll
    for (int g = 0; g < 8; ++g) {
      float v = acc2[g] + acc3[g] + biasb;
      v = v > 0.0f ? v : 0.0f;
      out[(size_t)(row0 + g + 8 * lhalf) * NOUT + ncol] = v;
    }
    __syncthreads();   // protect lds_t/lds_a before next tile
  }
}

// ---------------------------------------------------------------------------
extern "C" void kernel_launch(void* const* d_in, const int* in_sizes, int n_in,
                              void* d_out, int out_size, void* d_ws, size_t ws_size,
                              hipStream_t stream) {
  const float* x  = (const float*)d_in[0];
  const int*   ei = (const int*)d_in[1];   // (2, E): row 0 = src, row 1 = dst
  const float* W1 = (const float*)d_in[2];
  const float* b1 = (const float*)d_in[3];
  const float* W2 = (const float*)d_in[4];
  const float* b2 = (const float*)d_in[5];
  const float* W3 = (const float*)d_in[6];
  const float* b3 = (const float*)d_in[7];
  const float* b4w = (const float*)d_in[8]; // W4
  const float* b4 = (const float*)d_in[9];

  const int n_nodes = in_sizes[0] / N_FEAT;   // 100000
  const int n_edges = in_sizes[1] / 2;        // 1600000
  const int* srcI = ei;
  const int* dstI = ei + n_edges;

  float* buf0 = (float*)d_ws;                         // n_nodes x 128
  float* buf1 = buf0 + (size_t)n_nodes * N_FEAT;      // n_nodes x 128
  const size_t featBytes = (size_t)n_nodes * N_FEAT * sizeof(float);

  const int scatterThreads = 256;
  const int scatterBlocks = (int)(((long long)n_edges * 32 + scatterThreads - 1) / scatterThreads);

  // ---- layer 1 ----
  // buf0 = x  (so scatter-add yields x + agg directly)
  hipMemcpyAsync(buf0, x, featBytes, hipMemcpyDeviceToDevice, stream);
  gin_scatter_add<<<scatterBlocks, scatterThreads, 0, stream>>>(x, buf0, srcI, dstI, n_edges);
  // buf1 = relu(relu(buf0 @ W1 + b1) @ W2 + b2)
  gin_mlp<128, 128, 128><<<1024, 256, 0, stream>>>(buf0, W1, b1, W2, b2, buf1, n_nodes);

  // ---- layer 2 ----
  hipMemcpyAsync(buf0, buf1, featBytes, hipMemcpyDeviceToDevice, stream);
  gin_scatter_add<<<scatterBlocks, scatterThreads, 0, stream>>>(buf1, buf0, srcI, dstI, n_edges);
  // d_out = relu(relu(buf0 @ W3 + b3) @ W4 + b4)
  gin_mlp<128, 64, 64><<<1024, 128, 0, stream>>>(buf0, W3, b3, b4w, b4, (float*)d_out, n_nodes);
}